// Combination2_13486197309715
// MI455X (gfx1250) — compile-verified
//
#include <hip/hip_runtime.h>
#include <hip/hip_bf16.h>
#include <math.h>

#define NN 1024
#define BATCH 128
#define NPAIRS 523776            // N*(N-1)/2
#define B_TILE 32                // batch rows staged in LDS per block (128 KB)
#define THREADS 256              // 8 waves (wave32)
#define PPT 4                    // pairs per thread
#define P_TILE (THREADS * PPT)   // 1024 pairs per block

typedef __attribute__((ext_vector_type(4))) float v4f;
typedef __attribute__((ext_vector_type(4))) int   v4i;

__device__ __forceinline__ int row_offset(int i) {
    // number of pairs strictly before row i (triu, k=1, row-major)
    return (i * (2 * NN - 1 - i)) >> 1;
}

__global__ __launch_bounds__(THREADS) void pairlin_kernel(
    const float* __restrict__ x, const float* __restrict__ wk,
    float* __restrict__ out)
{
    __shared__ float xs[B_TILE * NN];  // 128 KB of the 320 KB WGP LDS

    const int tid = threadIdx.x;
    const int bbase = blockIdx.y * B_TILE;
    const int p0 = blockIdx.x * P_TILE + tid * PPT;

    // ---- Stage B_TILE rows of x into LDS ------------------------------
    const float* xg = x + (long long)bbase * NN;
#if defined(__gfx1250__) && __has_builtin(__builtin_amdgcn_global_load_async_to_lds_b128)
    {
        float* xgm = const_cast<float*>(xg);
        // 32768 floats / (256 lanes * 4 floats) = 32 async b128 per lane
        for (int c = 0; c < (B_TILE * NN) / (THREADS * 4); ++c) {
            int flat = (tid + c * THREADS) * 4;
            auto g = (__attribute__((address_space(1))) v4i*)(xgm + flat);
            auto l = (__attribute__((address_space(3))) v4i*)(&xs[flat]);
            __builtin_amdgcn_global_load_async_to_lds_b128(g, l, 0, 0);
        }
#if __has_builtin(__builtin_amdgcn_s_wait_asynccnt)
        __builtin_amdgcn_s_wait_asynccnt(0);
#else
        asm volatile("s_wait_asynccnt 0" ::: "memory");
#endif
    }
#else
    for (int f = tid; f < (B_TILE * NN) / 4; f += THREADS) {
        ((v4f*)xs)[f] = ((const v4f*)xg)[f];
    }
#endif
    __syncthreads();

    if (p0 >= NPAIRS) return;
    const int nvalid = min(PPT, NPAIRS - p0);

    // Prefetch this thread's weight pairs (global_prefetch_b8)
    __builtin_prefetch(wk + 2 * p0, 0, 0);

    // ---- Recover (i, j) from linear pair index p0 ---------------------
    // off(i) = i*(2N-1-i)/2 ; i = floor(((2N-1) - sqrt((2N-1)^2 - 8p))/2)
    const int M = 2 * NN - 1;
    long long disc = (long long)M * M - 8ll * p0;
    int i = (int)(((float)M - sqrtf((float)disc)) * 0.5f);
    if (i < 0) i = 0;
    if (i > NN - 2) i = NN - 2;
    while (i < NN - 2 && row_offset(i + 1) <= p0) ++i;  // fix float error
    while (i > 0 && row_offset(i) > p0) --i;
    int j = i + 1 + (p0 - row_offset(i));

    int ii[PPT], jj[PPT];
#pragma unroll
    for (int k = 0; k < PPT; ++k) {
        ii[k] = i;
        jj[k] = j;
        ++j;
        if (j >= NN) { ++i; j = i + 1; }
    }

    // ---- Load interleaved weights: kernel[2p]=w_i, kernel[2p+1]=w_j ---
    float wi[PPT], wj[PPT];
    if (nvalid == PPT) {
        const v4f* wp = (const v4f*)(wk + 2 * p0);  // 32B aligned (p0 % 4 == 0)
        v4f w01 = wp[0];
        v4f w23 = wp[1];
        wi[0] = w01.x; wj[0] = w01.y; wi[1] = w01.z; wj[1] = w01.w;
        wi[2] = w23.x; wj[2] = w23.y; wi[3] = w23.z; wj[3] = w23.w;
    } else {
#pragma unroll
        for (int k = 0; k < PPT; ++k) {
            int p = p0 + k;
            wi[k] = (k < nvalid) ? wk[2 * p] : 0.0f;
            wj[k] = (k < nvalid) ? wk[2 * p + 1] : 0.0f;
        }
    }

    // ---- Compute & stream non-temporal stores -------------------------
    if (nvalid == PPT) {
#pragma unroll 4
        for (int b = 0; b < B_TILE; ++b) {
            const float* xr = xs + b * NN;
            v4f o;
            o.x = fmaf(wi[0], xr[ii[0]], wj[0] * xr[jj[0]]);
            o.y = fmaf(wi[1], xr[ii[1]], wj[1] * xr[jj[1]]);
            o.z = fmaf(wi[2], xr[ii[2]], wj[2] * xr[jj[2]]);
            o.w = fmaf(wi[3], xr[ii[3]], wj[3] * xr[jj[3]]);
            v4f* dst = (v4f*)(out + (long long)(bbase + b) * NPAIRS + p0);
            __builtin_nontemporal_store(o, dst);  // NT: don't pollute L2
        }
    } else {
        for (int b = 0; b < B_TILE; ++b) {
            const float* xr = xs + b * NN;
            for (int k = 0; k < nvalid; ++k) {
                float o = fmaf(wi[k], xr[ii[k]], wj[k] * xr[jj[k]]);
                __builtin_nontemporal_store(
                    o, out + (long long)(bbase + b) * NPAIRS + p0 + k);
            }
        }
    }
}

extern "C" void kernel_launch(void* const* d_in, const int* in_sizes, int n_in,
                              void* d_out, int out_size, void* d_ws, size_t ws_size,
                              hipStream_t stream) {
    const float* x  = (const float*)d_in[0];   // (128, 1024) f32
    const float* wk = (const float*)d_in[1];   // (2*NUM_PAIRS,) f32 interleaved
    float* out = (float*)d_out;                // (128, NUM_PAIRS) f32

    dim3 grid((NPAIRS + P_TILE - 1) / P_TILE, BATCH / B_TILE, 1);  // 512 x 4
    pairlin_kernel<<<grid, THREADS, 0, stream>>>(x, wk, out);
}